// DownSample_58428735095310
// MI455X (gfx1250) — compile-verified
//
#include <hip/hip_runtime.h>
#include <cstdint>
#include <cstddef>

// GCNConv (project-first, PyG semantics) + BatchNorm1d (training stats), f32.
// N=100000 nodes, E=1.6M edges, IN_C=128, OUT_C=32.
//
// Pipeline (all on `stream`, graph-capture safe):
//   k_init     : deg=1 (self loop), zero stats, zero accumulator (d_out)
//   k_degree   : atomicAdd degree over edge targets
//   k_rsqrt    : deg -> deg^-1/2 in place
//   k_gemm     : h = x @ W via v_wmma_f32_16x16x4_f32 (one 16-row tile / wave,
//                W staged in LDS in K-pair-interleaved layout so each B
//                fragment is a single aligned ds_load_b64)
//   k_scatter  : msgs = dinv[r]*dinv[c] * h[r] scatter-added into acc (f32 atomics)
//   k_stats    : per-channel sum / sumsq of (acc + bias) -> 64-float scratch
//   k_finalize : y = (acc + b - mean) * rsqrt(var+eps) * gamma + beta -> d_out

#define IN_C   128
#define OUT_C  32
#define BN_EPS 1e-5f

typedef __attribute__((ext_vector_type(2))) float v2f;
typedef __attribute__((ext_vector_type(8))) float v8f;

// ---------------------------------------------------------------- init
__global__ void k_init(float* __restrict__ deg, float* __restrict__ stats,
                       float* __restrict__ acc, int N) {
  int i = blockIdx.x * blockDim.x + threadIdx.x;
  if (i < N * OUT_C) acc[i] = 0.0f;
  if (i < N)         deg[i] = 1.0f;   // self-loop contribution
  if (i < 64)        stats[i] = 0.0f;
}

// ---------------------------------------------------------------- degree
__global__ void k_degree(const int* __restrict__ dst, float* __restrict__ deg, int E) {
  int i = blockIdx.x * blockDim.x + threadIdx.x;
  if (i < E) atomicAdd(&deg[dst[i]], 1.0f);
}

__global__ void k_rsqrt(float* __restrict__ deg, int N) {
  int i = blockIdx.x * blockDim.x + threadIdx.x;
  if (i < N) {
    float d = deg[i];
    deg[i] = (d > 0.0f) ? rsqrtf(d) : 0.0f;
  }
}

// ---------------------------------------------------------------- GEMM: h = x @ W
// One wave computes one 16-row tile of h (16 x 32) as two 16x16 WMMA tiles.
// f32 WMMA 16x16x4 operand layout (ISA 7.12.2, 32-bit A 16x4):
//   A: lane l (half=l>>4, m=l&15): v0 = A[m][2*half], v1 = A[m][2*half+1]
//   B (4x16, mirrored):            v0 = B[2*half][n], v1 = B[2*half+1][n]
//   C/D: vgpr r: lanes 0-15 -> (M=r, N=lane), lanes 16-31 -> (M=r+8, N=lane-16)
//
// W is staged in LDS as K-pair-interleaved: sWp[(k>>1)*64 + n*2 + (k&1)],
// so the B fragment (W[k0][n], W[k0+1][n]) is one contiguous 8B-aligned pair
// -> single ds_load_b64 straight into an even VGPR pair (WMMA SRC1), no
// register shuffles; the two N-tiles are 128B apart (ds_load_2addr_b64-able).
__global__ void k_gemm(const float* __restrict__ x, const float* __restrict__ W,
                       float* __restrict__ h, int n_tiles) {
  __shared__ float sWp[IN_C * OUT_C];  // 16 KB, packed layout
  for (int i = threadIdx.x; i < IN_C * OUT_C; i += blockDim.x) {
    const int k = i >> 5;      // W is [IN_C][OUT_C] row-major
    const int n = i & 31;
    sWp[(k >> 1) * 64 + n * 2 + (k & 1)] = W[i];
  }
  __syncthreads();

  const int wave = threadIdx.x >> 5;
  const int lane = threadIdx.x & 31;
  const int tile = blockIdx.x * (blockDim.x >> 5) + wave;
  if (tile >= n_tiles) return;  // wave-uniform branch: EXEC stays all-ones for WMMA

  const int half = lane >> 4;   // 0: lanes 0-15, 1: lanes 16-31
  const int mn   = lane & 15;   // M index for A loads, N index for B loads

  const float* __restrict__ xrow = x + (size_t)(tile * 16 + mn) * IN_C;

  v8f c0 = {};  // N = 0..15
  v8f c1 = {};  // N = 16..31
  for (int k = 0; k < IN_C; k += 4) {
    const int k0 = k + 2 * half;              // this lane-half's K pair
    v2f a;
    a.x = xrow[k0];
    a.y = xrow[k0 + 1];
    const float* bp = &sWp[((k >> 1) + half) * 64 + mn * 2];
    v2f b0 = *(const v2f*)bp;                 // N-tile 0
    v2f b1 = *(const v2f*)(bp + 32);          // N-tile 1 (+128 B)
    c0 = __builtin_amdgcn_wmma_f32_16x16x4_f32(false, a, false, b0, (short)0, c0, false, false);
    c1 = __builtin_amdgcn_wmma_f32_16x16x4_f32(false, a, false, b1, (short)0, c1, false, false);
  }

  float* __restrict__ hbase = h + (size_t)tile * 16 * OUT_C;
  #pragma unroll
  for (int r = 0; r < 8; ++r) {
    const int m = r + 8 * half;
    hbase[(size_t)m * OUT_C + mn]      = c0[r];
    hbase[(size_t)m * OUT_C + 16 + mn] = c1[r];
  }
}

// ---------------------------------------------------------------- scatter-add
// i in [0, E+N): i < E -> real edge (src[i] -> dst[i]); i >= E -> self loop.
__global__ void k_scatter(const float* __restrict__ h, const int* __restrict__ src,
                          const int* __restrict__ dst, const float* __restrict__ dinv,
                          float* __restrict__ acc, int E, int N) {
  int i = blockIdx.x * blockDim.x + threadIdx.x;
  if (i >= E + N) return;
  int r, c;
  if (i < E) { r = src[i]; c = dst[i]; }
  else       { r = c = i - E; }
  const float nm = dinv[r] * dinv[c];
  const float4* __restrict__ hp = (const float4*)(h + (size_t)r * OUT_C);
  float* __restrict__ op = acc + (size_t)c * OUT_C;
  #pragma unroll
  for (int q = 0; q < OUT_C / 4; ++q) {
    float4 v = hp[q];
    atomicAdd(op + 4 * q + 0, nm * v.x);
    atomicAdd(op + 4 * q + 1, nm * v.y);
    atomicAdd(op + 4 * q + 2, nm * v.z);
    atomicAdd(op + 4 * q + 3, nm * v.w);
  }
}

// ---------------------------------------------------------------- BN statistics
// stats[0..31] = sum_c, stats[32..63] = sumsq_c over (acc + bias).
__global__ void k_stats(const float* __restrict__ acc, const float* __restrict__ bias,
                        float* __restrict__ stats, int N) {
  const int t = blockIdx.x * blockDim.x + threadIdx.x;
  const int c = t & (OUT_C - 1);
  const int rstart  = t >> 5;
  const int rstride = (gridDim.x * blockDim.x) >> 5;
  const float bc = bias[c];
  float s = 0.0f, s2 = 0.0f;
  for (int row = rstart; row < N; row += rstride) {
    float v = acc[(size_t)row * OUT_C + c] + bc;
    s  += v;
    s2 += v * v;
  }
  atomicAdd(&stats[c], s);
  atomicAdd(&stats[OUT_C + c], s2);
}

// ---------------------------------------------------------------- BN finalize
__global__ void k_finalize(float* __restrict__ acc, const float* __restrict__ bias,
                           const float* __restrict__ gamma, const float* __restrict__ beta,
                           const float* __restrict__ stats, int N) {
  int i = blockIdx.x * blockDim.x + threadIdx.x;
  if (i >= N * OUT_C) return;
  const int c = i & (OUT_C - 1);
  const float invN  = 1.0f / (float)N;
  const float mean  = stats[c] * invN;
  const float var   = stats[OUT_C + c] * invN - mean * mean;
  const float scale = rsqrtf(var + BN_EPS) * gamma[c];
  const float v = acc[i] + bias[c];
  acc[i] = (v - mean) * scale + beta[c];
}

// ---------------------------------------------------------------- launch
extern "C" void kernel_launch(void* const* d_in, const int* in_sizes, int n_in,
                              void* d_out, int out_size, void* d_ws, size_t ws_size,
                              hipStream_t stream) {
  const float* x     = (const float*)d_in[0];   // [N, 128]
  const int*   ei    = (const int*)  d_in[1];   // [2, E] (src row, dst row)
  const float* W     = (const float*)d_in[2];   // [128, 32]
  const float* b     = (const float*)d_in[3];   // [32]
  const float* gamma = (const float*)d_in[4];   // [32]
  const float* beta  = (const float*)d_in[5];   // [32]

  const int N = in_sizes[0] / IN_C;
  const int E = in_sizes[1] / 2;
  const int* src = ei;
  const int* dst = ei + E;

  // Workspace layout (floats): [deg/dinv: N][stats: 64][h: N*32]
  float* ws    = (float*)d_ws;
  float* deg   = ws;
  float* stats = ws + N;
  float* h     = ws + N + 64;          // byte offset (N+64)*4, 16B-aligned for N%4==0
  float* acc   = (float*)d_out;        // scatter accumulator, rewritten in-place by BN

  const int T = 256;
  const int n_tiles = (N + 15) / 16;               // 6250 (exact for N=100000)
  const int waves_per_block = T / 32;

  k_init    <<<(N * OUT_C + T - 1) / T, T, 0, stream>>>(deg, stats, acc, N);
  k_degree  <<<(E + T - 1) / T, T, 0, stream>>>(dst, deg, E);
  k_rsqrt   <<<(N + T - 1) / T, T, 0, stream>>>(deg, N);
  k_gemm    <<<(n_tiles + waves_per_block - 1) / waves_per_block, T, 0, stream>>>(x, W, h, n_tiles);
  k_scatter <<<(E + N + T - 1) / T, T, 0, stream>>>(h, src, dst, deg, acc, E, N);
  k_stats   <<<512, T, 0, stream>>>(acc, b, stats, N);
  k_finalize<<<(N * OUT_C + T - 1) / T, T, 0, stream>>>(acc, b, gamma, beta, stats, N);
}